// LagrangianDynamicalSystem_44092134261032
// MI455X (gfx1250) — compile-verified
//
#include <hip/hip_runtime.h>
#include <cmath>

// CDNA5 / gfx1250, wave32. One wave (32 threads) per workgroup, 16 samples per tile.
typedef float v2f __attribute__((ext_vector_type(2)));
typedef float v8f __attribute__((ext_vector_type(8)));

#define DIMQ   16
#define HID    128
#define NSAMP  16     // samples per workgroup tile
#define WROW   36     // padded row stride (floats) for W1 / y tiles (32 + 4; multiple of 4 for b128 stores)
#define SROW   130    // padded row stride for s / c1 / c2  ([sample][h]) — even for b64 loads
#define MROW   18     // padded row stride for augmented [M | rhs] rows (17 + 1)

// D = A(16x4 f32) * B(4x16 f32) + C(16x16 f32), wave32 WMMA
__device__ __forceinline__ v8f wmma4(v2f a, v2f b, v8f c) {
  // 8 args: (neg_a, A, neg_b, B, c_mod, C, reuse_a, reuse_b)
  return __builtin_amdgcn_wmma_f32_16x16x4_f32(false, a, false, b, (short)0, c, false, false);
}

// Branch-free tanh: tanh(x) = 1 - 2/(exp(2x)+1).
// v_exp_f32 is base-2: e^{2x} = exp2(x * 2*log2(e)). Saturates correctly:
// x->+inf: exp2->inf, rcp->0, t->1 ; x->-inf: exp2->0, t->-1.
__device__ __forceinline__ float fast_tanh(float x) {
  float e = __builtin_amdgcn_exp2f(x * 2.8853900817779268f);
  float r = __builtin_amdgcn_rcpf(e + 1.0f);
  return fmaf(-2.0f, r, 1.0f);
}

__global__ __launch_bounds__(32)
void lagr_dyn_kernel(const float* __restrict__ yg,
                     const float* __restrict__ w1g,
                     const float* __restrict__ b1g,
                     const float* __restrict__ w2g,
                     float* __restrict__ outg)
{
  __shared__ __align__(16) float w1_lds[HID * WROW];      // W1[h][j], padded
  __shared__ __align__(16) float y_lds [NSAMP * WROW];    // y tile [16][32], padded
  __shared__ __align__(16) float b1_lds[HID];
  __shared__ __align__(16) float w2_lds[HID];
  __shared__ __align__(16) float s_lds [NSAMP * SROW];    // s  [sample][h]
  __shared__ __align__(16) float c1_lds[NSAMP * SROW];    // u  [sample][h]
  __shared__ __align__(16) float c2_lds[NSAMP * SROW];    // -s*a [sample][h]
  __shared__ __align__(16) float m_lds [NSAMP * 16 * MROW]; // per-sample [M | rhs] augmented rows

  const int lane = threadIdx.x;         // 0..31 (wave32)
  const int lo   = lane & 15;
  const int hi8  = (lane & 16) >> 1;    // 0 (lanes 0-15) or 8 (lanes 16-31): C-layout row offset
  const int d0   = (lane & 16) >> 3;    // 0 or 2: A/B-layout K offset for the lane half
  const int tile = blockIdx.x;          // 16 samples per tile

  // ------------------------------------------------------------------
  // Stage constants + y tile into LDS (b128 global loads, b128 LDS stores)
  // ------------------------------------------------------------------
  {
    const float4* g = (const float4*)w1g;                 // 128x32 floats = 1024 float4
    for (int i = lane; i < (HID * 32) / 4; i += 32) {
      float4 v = g[i];
      int row = i >> 3, col = (i & 7) << 2;
      *(float4*)&w1_lds[row * WROW + col] = v;
    }
    const float4* yg4 = (const float4*)(yg + (size_t)tile * NSAMP * 32); // 16x32 = 128 float4
    for (int i = lane; i < (NSAMP * 32) / 4; i += 32) {
      float4 v = yg4[i];
      int row = i >> 3, col = (i & 7) << 2;
      *(float4*)&y_lds[row * WROW + col] = v;
    }
    float4 bv = ((const float4*)b1g)[lane];
    *(float4*)&b1_lds[lane << 2] = bv;
    float4 wv = ((const float4*)w2g)[lane];
    *(float4*)&w2_lds[lane << 2] = wv;
  }
  // Single-wave workgroup: DS ops are in-order within the wave — no barrier needed.

  // ------------------------------------------------------------------
  // Phase 1: Z = Y*W1^T + b1 ; a = Qdot*W1a^T ; u, s, c2 -> LDS
  // C layout: element r <-> sample (r+hi8), column h = hb*16 + lo
  // ------------------------------------------------------------------
  for (int hb = 0; hb < 8; ++hb) {
    const int hrow = hb * 16 + lo;
    v8f zc = {};
    #pragma unroll
    for (int c = 0; c < 8; ++c) {                 // K = 32 over y columns
      v2f a = *(const v2f*)&y_lds [lo * WROW + 4 * c + d0];
      v2f b = *(const v2f*)&w1_lds[hrow * WROW + 4 * c + d0];
      zc = wmma4(a, b, zc);
    }
    v8f ac = {};
    #pragma unroll
    for (int c = 0; c < 4; ++c) {                 // a_h = sum_i qdot_i * W1[h][i], K = 16 (q-columns!)
      v2f a = *(const v2f*)&y_lds [lo * WROW + 16 + 4 * c + d0];
      v2f b = *(const v2f*)&w1_lds[hrow * WROW + 4 * c + d0];
      ac = wmma4(a, b, ac);
    }
    const float bb = b1_lds[hrow];
    const float ww = w2_lds[hrow];
    #pragma unroll
    for (int r = 0; r < 8; ++r) {
      float z = zc[r] + bb;
      float t = fast_tanh(z);                     // branch-free: keeps EXEC all-ones
      float u = ww * (1.0f - t * t);              // w2*(1-t^2)
      float s = -2.0f * t * u;                    // -2*w2*t*(1-t^2)
      int off = (r + hi8) * SROW + hrow;
      c1_lds[off] = u;
      s_lds [off] = s;
      c2_lds[off] = -s * ac[r];                   // -s*a
    }
  }

  // ------------------------------------------------------------------
  // Phase 2: per-sample M = I + W1b^T diag(s) W1b  (K = 128, 32 WMMA/sample)
  // chunk-outer / sample-inner: A chunk (constant) reused across 16 samples,
  // B chunk = A values * per-sample s pair (single ds_load_b64).
  // ------------------------------------------------------------------
  v8f acc[NSAMP];
  #pragma unroll
  for (int b = 0; b < NSAMP; ++b) {
    v8f c;
    #pragma unroll
    for (int r = 0; r < 8; ++r) c[r] = (lo == r + hi8) ? 1.0f : 0.0f;  // identity
    acc[b] = c;
  }
  for (int c = 0; c < 32; ++c) {
    const int h0 = 4 * c + d0;
    float a0 = w1_lds[h0 * WROW + 16 + lo];       // W1b[h0]  [16+lo]
    float a1 = w1_lds[(h0 + 1) * WROW + 16 + lo]; // W1b[h0+1][16+lo]
    v2f av; av.x = a0; av.y = a1;
    #pragma unroll
    for (int b = 0; b < NSAMP; ++b) {
      v2f sv = *(const v2f*)&s_lds[b * SROW + h0];
      v2f bv; bv.x = a0 * sv.x; bv.y = a1 * sv.y;
      acc[b] = wmma4(av, bv, acc[b]);
    }
  }
  #pragma unroll
  for (int b = 0; b < NSAMP; ++b) {
    #pragma unroll
    for (int r = 0; r < 8; ++r)
      m_lds[(b * 16 + r + hi8) * MROW + lo] = acc[b][r];
  }

  // ------------------------------------------------------------------
  // Phase 3: rhs = C1*W1a + C2*W1b  ([16 x 128] @ [128 x 16], 64 WMMA)
  // ------------------------------------------------------------------
  v8f rc = {};
  for (int c = 0; c < 32; ++c) {
    const int h0 = 4 * c + d0;
    v2f a1v = *(const v2f*)&c1_lds[lo * SROW + h0];
    v2f a2v = *(const v2f*)&c2_lds[lo * SROW + h0];
    v2f b1v; b1v.x = w1_lds[h0 * WROW + lo];       b1v.y = w1_lds[(h0 + 1) * WROW + lo];
    v2f b2v; b2v.x = w1_lds[h0 * WROW + 16 + lo];  b2v.y = w1_lds[(h0 + 1) * WROW + 16 + lo];
    rc = wmma4(a1v, b1v, rc);
    rc = wmma4(a2v, b2v, rc);
  }
  #pragma unroll
  for (int r = 0; r < 8; ++r)
    m_lds[((r + hi8) * 16 + lo) * MROW + 16] = rc[r];   // augmented column

  // ------------------------------------------------------------------
  // Phase 4: solve M x = rhs, Gauss-Jordan (no pivoting: M ~ I + small).
  // 2 samples per pass; lane owns one row of [M | rhs]; pivot-row broadcast
  // via __shfl (ds_bpermute). x_j = final r[16].
  // ------------------------------------------------------------------
  const int base = lane & 16;
  for (int p = 0; p < 8; ++p) {
    const int b = 2 * p + (base >> 4);
    float r[17];
    #pragma unroll
    for (int m = 0; m < 17; ++m) r[m] = m_lds[(b * 16 + lo) * MROW + m];
    #pragma unroll
    for (int k = 0; k < 16; ++k) {
      float piv = __shfl(r[k], base + k, 32);
      float inv = __builtin_amdgcn_rcpf(piv);
      inv = inv * fmaf(-piv, inv, 2.0f);          // one Newton step: ~0.5 ulp
      float fac = r[k] * inv;
      bool mine = (lo == k);
      #pragma unroll
      for (int m = k + 1; m < 17; ++m) {
        float pv = __shfl(r[m], base + k, 32);
        r[m] = mine ? (r[m] * inv) : fmaf(-fac, pv, r[m]);
      }
    }
    outg[(size_t)(tile * NSAMP + b) * DIMQ + lo] = r[16];
  }
}

extern "C" void kernel_launch(void* const* d_in, const int* in_sizes, int n_in,
                              void* d_out, int out_size, void* d_ws, size_t ws_size,
                              hipStream_t stream) {
  const float* y  = (const float*)d_in[0];   // [BATCH, 32]
  const float* W1 = (const float*)d_in[1];   // [128, 32]
  const float* b1 = (const float*)d_in[2];   // [128]
  const float* w2 = (const float*)d_in[3];   // [128]
  float* out = (float*)d_out;                // [BATCH, 16]
  (void)n_in; (void)out_size; (void)d_ws; (void)ws_size;
  const int batch  = in_sizes[0] / 32;       // 32768
  const int blocks = batch / NSAMP;          // 2048 tiles, 16 samples each
  lagr_dyn_kernel<<<blocks, 32, 0, stream>>>(y, W1, b1, w2, out);
}